// Advection3D_25082609009398
// MI455X (gfx1250) — compile-verified
//
#include <hip/hip_runtime.h>
#include <hip/hip_bf16.h>

// Problem constants (match reference)
#define NZ 32
#define NY 1024
#define NX 1024
#define TX 64
#define TY 4
#define SH_W (TX + 6)   // 70
#define SH_H (TY + 6)   // 10
#define NFE (TY * (TX + 1))        // 260 x-face fluxes per tile
#define NFN ((TY + 1) * TX)        // 320 y-face fluxes per tile

__device__ __forceinline__ int iclamp(int x, int lo, int hi) {
    return x < lo ? lo : (x > hi ? hi : x);
}

// Async global->LDS copy of one dword (gfx1250 GLOBAL_LOAD_ASYNC_TO_LDS_B32,
// tracked by ASYNCcnt). LDS destination address = low 32 bits of the generic
// pointer.
__device__ __forceinline__ void async_copy_b32(float* lds_dst, const float* gsrc) {
    unsigned lds_addr = (unsigned)(unsigned long long)lds_dst;
    asm volatile("global_load_async_to_lds_b32 %0, %1, off"
                 :: "v"(lds_addr), "v"(gsrc)
                 : "memory");
}

__device__ __forceinline__ void wait_async_all() {
    asm volatile("s_wait_asynccnt 0" ::: "memory");
}

// Classic Jiang-Shu WENO5 reconstruction at the i+1/2 face (left-biased).
__device__ __forceinline__ float weno5_recon(float qm2, float qm1, float q0,
                                             float qp1, float qp2) {
    const float c1312 = 13.0f / 12.0f;
    float d0 = qm2 - 2.0f * qm1 + q0;
    float e0 = qm2 - 4.0f * qm1 + 3.0f * q0;
    float b0 = c1312 * d0 * d0 + 0.25f * e0 * e0;
    float d1 = qm1 - 2.0f * q0 + qp1;
    float e1 = qm1 - qp1;
    float b1 = c1312 * d1 * d1 + 0.25f * e1 * e1;
    float d2 = q0 - 2.0f * qp1 + qp2;
    float e2 = 3.0f * q0 - 4.0f * qp1 + qp2;
    float b2 = c1312 * d2 * d2 + 0.25f * e2 * e2;
    float t0 = __builtin_amdgcn_rcpf(1e-6f + b0);
    float t1 = __builtin_amdgcn_rcpf(1e-6f + b1);
    float t2 = __builtin_amdgcn_rcpf(1e-6f + b2);
    float a0 = 0.1f * t0 * t0;
    float a1 = 0.6f * t1 * t1;
    float a2 = 0.3f * t2 * t2;
    const float sixth = 1.0f / 6.0f;
    float p0 = (2.0f * qm2 - 7.0f * qm1 + 11.0f * q0) * sixth;
    float p1 = (-qm1 + 5.0f * q0 + 2.0f * qp1) * sixth;
    float p2 = (2.0f * q0 + 5.0f * qp1 - qp2) * sixth;
    return (a0 * p0 + a1 * p1 + a2 * p2) * __builtin_amdgcn_rcpf(a0 + a1 + a2);
}

// Upwinded flux at a face. s points at q[i-2]; stride between stencil samples.
// Key optimization: where(vel>=0, recon(qm2..qp2), recon(qp3..qm1)) is the SAME
// reconstruction applied to a mirrored stencil, so select the 5 inputs per lane
// (5 v_cndmask) and run ONE reconstruction instead of two.
__device__ __forceinline__ float weno5_flux(const float* s, int stride, float vel) {
    float q0 = s[0 * stride];   // q[i-2]
    float q1 = s[1 * stride];   // q[i-1]
    float q2 = s[2 * stride];   // q[i]
    float q3 = s[3 * stride];   // q[i+1]
    float q4 = s[4 * stride];   // q[i+2]
    float q5 = s[5 * stride];   // q[i+3]
    bool up = (vel >= 0.0f);
    float m2 = up ? q0 : q5;
    float m1 = up ? q1 : q4;
    float c0 = up ? q2 : q3;
    float p1 = up ? q3 : q2;
    float p2 = up ? q4 : q1;
    return vel * weno5_recon(m2, m1, c0, p1, p2);
}

__global__ __launch_bounds__(256)
void weno5_advect_kernel(const float* __restrict__ h,
                         const float* __restrict__ u,
                         const float* __restrict__ v,
                         float* __restrict__ out) {
    __shared__ float sh[SH_H * SH_W];        // haloed h tile
    __shared__ float sfe[TY][TX + 1];        // x-face fluxes fe[x0-1 .. x0+TX-1]
    __shared__ float sfn[TY + 1][TX];        // y-face fluxes fn[y0-1 .. y0+TY-1]

    const int tid = threadIdx.x;
    const int x0 = blockIdx.x * TX;
    const int y0 = blockIdx.y * TY;
    const int z  = blockIdx.z;

    const int tx = tid % TX;
    const int ty = tid / TX;
    const int gx = x0 + tx;
    const int gy = y0 + ty;
    const size_t plane = (size_t)z * NY;
    const size_t out_idx = (plane + gy) * NX + gx;

    // z boundary planes are all zeros (uniform branch within a block).
    if (z == 0 || z == NZ - 1) {
        __builtin_nontemporal_store(0.0f, &out[out_idx]);
        return;
    }

    // Stage 1: async-copy haloed h tile into LDS with edge clamping
    // (clamped addresses implement jnp.pad(mode='edge') at domain borders).
    for (int i = tid; i < SH_H * SH_W; i += 256) {
        int r = i / SH_W;
        int c = i % SH_W;
        int sy = iclamp(y0 + r - 3, 0, NY - 1);
        int sx = iclamp(x0 + c - 3, 0, NX - 1);
        async_copy_b32(&sh[i], h + (plane + sy) * NX + sx);
    }
    wait_async_all();
    __syncthreads();

    // Stage 2: one WENO5 flux per face, shared between neighbor cells.
    // u/v are single-touch streams -> non-temporal loads keep L2 for h.
    for (int j = tid; j < NFE + NFN; j += 256) {
        if (j < NFE) {
            // fe at global x-face index xf = x0 - 1 + fx, row gy2 = y0 + tr
            int tr = j / (TX + 1);
            int fx = j % (TX + 1);
            int xf = iclamp(x0 - 1 + fx, 0, NX - 1);   // clamp only matters at edge tiles
            float vel = __builtin_nontemporal_load(&u[(plane + (y0 + tr)) * NX + xf]);
            // stencil h[xf-2 .. xf+3] -> LDS row tr+3, cols fx .. fx+5
            sfe[tr][fx] = weno5_flux(&sh[(tr + 3) * SH_W + fx], 1, vel);
        } else {
            // fn at global y-face index yf = y0 - 1 + fy, col gx2 = x0 + cx
            int k  = j - NFE;
            int fy = k / TX;
            int cx = k % TX;
            int yf = iclamp(y0 - 1 + fy, 0, NY - 1);
            float vel = __builtin_nontemporal_load(&v[(plane + yf) * NX + (x0 + cx)]);
            // stencil h[yf-2 .. yf+3] -> LDS rows fy .. fy+5, col cx+3
            sfn[fy][cx] = weno5_flux(&sh[fy * SH_W + (cx + 3)], SH_W, vel);
        }
    }
    __syncthreads();

    // Stage 3: divergence from shared faces; zero on x/y boundary ring.
    float val = 0.0f;
    if (gx >= 2 && gx <= NX - 3 && gy >= 2 && gy <= NY - 3) {
        float div = (sfe[ty][tx + 1] - sfe[ty][tx]) * (1.0f / 1000.0f)
                  + (sfn[ty + 1][tx] - sfn[ty][tx]) * (1.0f / 1000.0f);
        val = -div;
    }
    __builtin_nontemporal_store(val, &out[out_idx]);
}

extern "C" void kernel_launch(void* const* d_in, const int* in_sizes, int n_in,
                              void* d_out, int out_size, void* d_ws, size_t ws_size,
                              hipStream_t stream) {
    (void)in_sizes; (void)n_in; (void)out_size; (void)d_ws; (void)ws_size;
    const float* h = (const float*)d_in[0];
    const float* u = (const float*)d_in[1];
    const float* v = (const float*)d_in[2];
    float* out = (float*)d_out;

    dim3 grid(NX / TX, NY / TY, NZ);   // 16 x 256 x 32
    dim3 block(256, 1, 1);             // 8 wave32 per workgroup
    weno5_advect_kernel<<<grid, block, 0, stream>>>(h, u, v, out);
}